// Cross_Decoder_84181359002211
// MI455X (gfx1250) — compile-verified
//
#include <hip/hip_runtime.h>

typedef float v2f __attribute__((ext_vector_type(2)));
typedef float v8f __attribute__((ext_vector_type(8)));

#define IN_FEAT 256
#define OUT_FEAT 32

// ---------------------------------------------------------------------------
// Stage 0: zero the output accumulator (harness poisons d_out with 0xAA).
// ---------------------------------------------------------------------------
__global__ void zero_kernel(float* __restrict__ p, int n) {
    int i = blockIdx.x * blockDim.x + threadIdx.x;
    if (i < n) p[i] = 0.0f;
}

// ---------------------------------------------------------------------------
// Stage 1: dense projection x = feat @ W using V_WMMA_F32_16X16X4_F32.
// Block = 256 threads = 8 waves; each wave computes a 32-row x 32-col tile
// (2 A fragments x 2 B fragments -> 4 WMMAs per K-step of 4).
//
// W is staged in LDS *pair-interleaved*: sWp[(k>>1)*64 + n*2 + (k&1)],
// so a B fragment (two consecutive K values for one N) is a single
// contiguous 8-byte LDS read into an aligned VGPR pair, and the two N-tiles
// (n, n+16) are 128 B apart -> one ds_load_2addr_b64.
//
// fp32 WMMA fragment layouts (ISA 7.12.2):
//   A 16x4 : 2 VGPRs; lane&15 = M; lanes 0-15 -> K = {k0, k0+1},
//                                   lanes 16-31 -> K = {k0+2, k0+3}
//   B 4x16 : 2 VGPRs; lane&15 = N; same K split as A
//   C/D    : 8 VGPRs; VGPR r -> M = r + 8*(lane>=16); N = lane&15
// ---------------------------------------------------------------------------
__global__ __launch_bounds__(256) void gemm_proj_kernel(
        const float* __restrict__ feat,   // [nNodes, 256]
        const float* __restrict__ W,      // [256, 32]
        float* __restrict__ x,            // [nNodes, 32]
        int nNodes) {
    __shared__ float sWp[IN_FEAT * OUT_FEAT];   // 32 KB, pair-interleaved

    // Cooperative load + pair-interleave of W.
    for (int i = threadIdx.x; i < (IN_FEAT * OUT_FEAT) / 4; i += 256) {
        float4 w4 = ((const float4*)W)[i];     // W[k][n..n+3]
        int k = i >> 3;                        // (i*4) / 32
        int n = (i & 7) * 4;                   // (i*4) % 32
        float* d = &sWp[(k >> 1) * 64 + (k & 1)];
        d[(n + 0) * 2] = w4.x;
        d[(n + 1) * 2] = w4.y;
        d[(n + 2) * 2] = w4.z;
        d[(n + 3) * 2] = w4.w;
    }
    __syncthreads();

    const int lane = threadIdx.x & 31;
    const int wave = threadIdx.x >> 5;
    const int m0   = (blockIdx.x * 8 + wave) * 32;
    if (m0 >= nNodes) return;   // wave-uniform exit

    const int mrow = lane & 15;
    const int ncol = lane & 15;
    const int kk   = (lane >> 4) * 2;   // 0 for lanes 0-15, 2 for lanes 16-31

    // Clamp A-row indices for the tail tile (WMMA needs EXEC all-ones, so
    // clamp addresses instead of predicating loads).
    int arow0 = m0 + mrow;       if (arow0 >= nNodes) arow0 = nNodes - 1;
    int arow1 = m0 + 16 + mrow;  if (arow1 >= nNodes) arow1 = nNodes - 1;
    const v2f* __restrict__ ap0 =
        (const v2f*)(feat + (size_t)arow0 * IN_FEAT + kk);   // 8B aligned
    const v2f* __restrict__ ap1 =
        (const v2f*)(feat + (size_t)arow1 * IN_FEAT + kk);

    // B base: pair-row (kk>>1), column ncol.
    const float* __restrict__ bbase = sWp + (kk >> 1) * 64 + ncol * 2;

    v8f acc00 = {};   // rows m0..m0+15,   cols 0..15
    v8f acc01 = {};   // rows m0..m0+15,   cols 16..31
    v8f acc10 = {};   // rows m0+16..+31,  cols 0..15
    v8f acc11 = {};   // rows m0+16..+31,  cols 16..31

    #pragma unroll 4
    for (int k0 = 0; k0 < IN_FEAT; k0 += 4) {
        // feat is streamed exactly once -> non-temporal, keep L2 free for x.
        v2f a0 = __builtin_nontemporal_load(&ap0[k0 >> 1]);
        v2f a1 = __builtin_nontemporal_load(&ap1[k0 >> 1]);

        const v2f* bp = (const v2f*)(bbase + (k0 >> 1) * 64);
        v2f b0 = bp[0];    // cols ncol
        v2f b1 = bp[16];   // cols ncol+16 (128 B apart -> ds_load_2addr_b64)

        acc00 = __builtin_amdgcn_wmma_f32_16x16x4_f32(
                    false, a0, false, b0, (short)0, acc00, false, false);
        acc01 = __builtin_amdgcn_wmma_f32_16x16x4_f32(
                    false, a0, false, b1, (short)0, acc01, false, false);
        acc10 = __builtin_amdgcn_wmma_f32_16x16x4_f32(
                    false, a1, false, b0, (short)0, acc10, false, false);
        acc11 = __builtin_amdgcn_wmma_f32_16x16x4_f32(
                    false, a1, false, b1, (short)0, acc11, false, false);
    }

    // Scatter D: VGPR r -> row (tile + r + 8*(lane>=16)), col = lane&15.
    const int msub = (lane >> 4) * 8;
    float* __restrict__ xr0 = x + (size_t)(m0 + msub) * OUT_FEAT + ncol;
    float* __restrict__ xr1 = xr0 + 16 * OUT_FEAT;

    if (m0 + 32 <= nNodes) {
        // Fast path: whole 32-row tile in range, no per-row predication.
        #pragma unroll
        for (int r = 0; r < 8; ++r) {
            xr0[r * OUT_FEAT]      = acc00[r];
            xr0[r * OUT_FEAT + 16] = acc01[r];
            xr1[r * OUT_FEAT]      = acc10[r];
            xr1[r * OUT_FEAT + 16] = acc11[r];
        }
    } else {
        // Tail tile: guard each row.
        #pragma unroll
        for (int r = 0; r < 8; ++r) {
            if (m0 + msub + r < nNodes) {
                xr0[r * OUT_FEAT]      = acc00[r];
                xr0[r * OUT_FEAT + 16] = acc01[r];
            }
            if (m0 + 16 + msub + r < nNodes) {
                xr1[r * OUT_FEAT]      = acc10[r];
                xr1[r * OUT_FEAT + 16] = acc11[r];
            }
        }
    }
}

// ---------------------------------------------------------------------------
// Stage 2: COO SpMM, out[row[e]] += vals[e] * x[col[e]].
// One wave per edge; lane f handles feature f. x (12.8 MB) lives in L2, so
// the gather and the 128 B atomic-add burst both stay on-chip.
// ---------------------------------------------------------------------------
__global__ __launch_bounds__(256) void spmm_kernel(
        const float* __restrict__ x,        // [nNodes, 32]
        const float* __restrict__ vals,     // [nEdges]
        const int*   __restrict__ erow,     // [nEdges]
        const int*   __restrict__ ecol,     // [nEdges]
        float* __restrict__ out,            // [nNodes, 32]
        int nEdges) {
    const int e    = (blockIdx.x * blockDim.x + threadIdx.x) >> 5;
    const int lane = threadIdx.x & 31;
    if (e >= nEdges) return;

    // Edge streams are read once per pass -> non-temporal.
    const int   r = __builtin_nontemporal_load(&erow[e]);
    const int   c = __builtin_nontemporal_load(&ecol[e]);
    const float v = __builtin_nontemporal_load(&vals[e]);

    const float msg = v * x[(size_t)c * OUT_FEAT + lane];
    atomicAdd(&out[(size_t)r * OUT_FEAT + lane], msg);
}

// ---------------------------------------------------------------------------
// Launcher
// ---------------------------------------------------------------------------
extern "C" void kernel_launch(void* const* d_in, const int* in_sizes, int n_in,
                              void* d_out, int out_size, void* d_ws, size_t ws_size,
                              hipStream_t stream) {
    const float* feat = (const float*)d_in[0];   // [N, 256] f32
    const float* W    = (const float*)d_in[1];   // [256, 32] f32
    const float* vals = (const float*)d_in[2];   // [E] f32
    const int*   erow = (const int*)d_in[3];     // [E] i32
    const int*   ecol = (const int*)d_in[4];     // [E] i32
    float*       out  = (float*)d_out;           // [N, 32] f32

    const int nNodes = in_sizes[0] / IN_FEAT;
    const int nEdges = in_sizes[2];

    float* x = (float*)d_ws;                     // [N, 32] f32 scratch (12.8 MB)

    // 1) zero output
    zero_kernel<<<(out_size + 255) / 256, 256, 0, stream>>>(out, out_size);

    // 2) projection GEMM (WMMA fp32): 8 waves/block, 32 rows per wave
    const int mtiles = (nNodes + 255) / 256;
    gemm_proj_kernel<<<mtiles, 256, 0, stream>>>(feat, W, x, nNodes);

    // 3) edge-parallel SpMM with fp32 atomics (1 wave per edge)
    const long long threadsNeeded = (long long)nEdges * 32;
    const int blocks = (int)((threadsNeeded + 255) / 256);
    spmm_kernel<<<blocks, 256, 0, stream>>>(x, vals, erow, ecol, out, nEdges);
}